// PositionalEncoding_30915174597069
// MI455X (gfx1250) — compile-verified
//
#include <hip/hip_runtime.h>
#include <stdint.h>

// ---------------------------------------------------------------------------
// PositionalEncoding add:  out[b,s,:] = x[b,s,:] + pix(s) + grd(s) + pr(s)
// Streaming kernel: 256MB in + 256MB out => ~22us at 23.3 TB/s HBM.
// CDNA5 paths: GLOBAL_LOAD_ASYNC_TO_LDS_B128 (ASYNCcnt) staging of the
// pixels_pe row, s_wait_asynccnt, non-temporal B128 loads/stores for the
// big x/out streams (keeps the <2MB PE tables resident in L2/WGP$).
// Segment lookup: s is wave-uniform (readfirstlane), so the searchsorted
// select-scan runs entirely in SALU off s_load_b512'd start tables.
// ---------------------------------------------------------------------------

typedef float f4 __attribute__((ext_vector_type(4)));

static constexpr int kD         = 512;     // D_MODEL
static constexpr int kB         = 8;       // batch
static constexpr int kS         = 16384;   // sequence
static constexpr int kMaxPixels = 900;
static constexpr int kMaxPairs  = 16;
static constexpr int kNGrids    = 32;
static constexpr int kNPairs    = 16;
static constexpr int kD4        = kD / 4;  // 128 float4 per row
static constexpr int kRows      = 2;       // positions per block

__global__ __launch_bounds__(kRows * kD4) void pe_add_kernel(
    const float* __restrict__ x,
    const float* __restrict__ pixels_pe,
    const float* __restrict__ grids_pe,
    const float* __restrict__ pairs_pe,
    const int*   __restrict__ grid_starts,
    const int*   __restrict__ grid_lengths,
    const int*   __restrict__ pair_starts,
    const int*   __restrict__ pair_lengths,
    float* __restrict__ out)
{
    __shared__ float pixrow[kRows * kD];          // 4 KB staging for pixels_pe rows

    const int t = threadIdx.x;
    const int r = t >> 7;                         // row within block: 0..1
    const int c = t & (kD4 - 1);                  // float4 column: 0..127
    // s is identical across every wave32 (waves are 32 contiguous threads,
    // all with the same r) -> force it scalar so the segment lookup lowers
    // to pure SALU over the s_load'ed start tables.
    const int s = __builtin_amdgcn_readfirstlane(blockIdx.x * kRows + r);

    // ---- searchsorted(grid_starts, s, 'right')-1 as an unrolled select-scan
    //      (all-scalar: s_load tables + s_cmp/s_cselect chain) ----
    int gid = 0;
    int gstart = grid_starts[0];
#pragma unroll
    for (int i = 1; i < kNGrids; ++i) {
        const int st = grid_starts[i];
        const bool le = (st <= s);
        gid    = le ? i  : gid;
        gstart = le ? st : gstart;
    }
    const int  goff  = s - gstart;
    const bool gmask = goff < grid_lengths[gid];
    int gclip = goff < 0 ? 0 : goff;
    if (gclip > kMaxPixels - 1) gclip = kMaxPixels - 1;

    int pid = 0;
    int pstart = pair_starts[0];
#pragma unroll
    for (int i = 1; i < kNPairs; ++i) {
        const int st = pair_starts[i];
        const bool le = (st <= s);
        pid    = le ? i  : pid;
        pstart = le ? st : pstart;
    }
    const int  poff  = s - pstart;
    const bool pmask = poff < pair_lengths[pid];
    int pclip = pid;
    if (pclip > kMaxPairs - 1) pclip = kMaxPairs - 1;

    // ---- async-stage pixels_pe row into LDS (each lane copies its own 16B).
    //      GVS mode: mem_addr = SADDR(64) + VADDR(32); LDS dest addr in VDST.
    {
        const unsigned lds_dst = (unsigned)(uintptr_t)(&pixrow[r * kD + c * 4]);
        const unsigned gbytes  = (unsigned)gclip * (unsigned)(kD * sizeof(float))
                               + (unsigned)c * 16u;
        asm volatile(
            "global_load_async_to_lds_b128 %0, %1, %2"
            :
            : "v"(lds_dst), "v"(gbytes), "s"(pixels_pe)
            : "memory");
    }

    // ---- issue ALL long-latency work before waiting on the async copy:
    //      tiny grids/pairs rows (L2/WGP$ resident) + the 8 NT x loads ----
    const f4* __restrict__ g4 = (const f4*)(grids_pe + (size_t)(gid & 1) * kD);
    const f4* __restrict__ p4 = (const f4*)(pairs_pe + (size_t)pclip * kD);
    f4 grd = gmask ? g4[c] : (f4)(0.0f);
    f4 pr  = pmask ? p4[c] : (f4)(0.0f);

    const f4* __restrict__ x4 = (const f4*)x;
    f4*       __restrict__ o4 = (f4*)out;
    const size_t rowbase = (size_t)s * kD4 + c;

    f4 v[kB];
#pragma unroll
    for (int b = 0; b < kB; ++b) {
        v[b] = __builtin_nontemporal_load(&x4[(size_t)b * kS * kD4 + rowbase]);
    }

    // ---- now wait for the async PE staging; each lane reads back exactly
    //      the bytes it staged itself, so no workgroup barrier is needed ----
    asm volatile("s_wait_asynccnt 0" ::: "memory");
    const f4* __restrict__ lp = (const f4*)(&pixrow[r * kD]);
    f4 pix = lp[c];
    if (!gmask) pix = (f4)(0.0f);

    const f4 comb = pix + grd + pr;

    // ---- combined row reused for all 8 batch rows; NT stores bypass cache ----
#pragma unroll
    for (int b = 0; b < kB; ++b) {
        f4 o = v[b] + comb;
        __builtin_nontemporal_store(o, &o4[(size_t)b * kS * kD4 + rowbase]);
    }
}

extern "C" void kernel_launch(void* const* d_in, const int* in_sizes, int n_in,
                              void* d_out, int out_size, void* d_ws, size_t ws_size,
                              hipStream_t stream) {
    const float* x            = (const float*)d_in[0];
    const float* pixels_pe    = (const float*)d_in[1];
    const float* grids_pe     = (const float*)d_in[2];
    const float* pairs_pe     = (const float*)d_in[3];
    const int*   grid_starts  = (const int*)d_in[4];
    const int*   grid_lengths = (const int*)d_in[5];
    const int*   pair_starts  = (const int*)d_in[6];
    const int*   pair_lengths = (const int*)d_in[7];
    float*       out          = (float*)d_out;

    dim3 grid(kS / kRows);          // 8192 blocks
    dim3 block(kRows * kD4);        // 256 threads = 8 wave32

    hipLaunchKernelGGL(pe_add_kernel, grid, block, 0, stream,
                       x, pixels_pe, grids_pe, pairs_pe,
                       grid_starts, grid_lengths, pair_starts, pair_lengths, out);
}